// SEPGCN_66743791780380
// MI455X (gfx1250) — compile-verified
//
#include <hip/hip_runtime.h>
#include <hip/hip_bf16.h>

// ---------------- problem constants (match reference) ----------------
#define NUM_USERS  50000
#define NUM_ITEMS  50000
#define NUM_CATS   500
#define LATENT_DIM 64
#define N_LAYERS   3
#define N_EDGES    500000
#define NNZ_G      1000000
#define NNZ_GX     2000000
#define BATCH_SZ   8192
#define N_NODES    (NUM_USERS + NUM_ITEMS)
#define NODE_ELEMS (N_NODES * LATENT_DIM)        // 6,400,000 floats
#define EDGE_FEATS (N_EDGES * 2 * LATENT_DIM)    // 64,000,000 floats

typedef __attribute__((ext_vector_type(2))) float v2f;
typedef __attribute__((ext_vector_type(8))) float v8f;

__device__ __forceinline__ void atomAddF(float* p, float v) {
    unsafeAtomicAdd(p, v);   // hardware global_atomic_add_f32 on gfx1250
}

// ---------------- init: all_emb = concat(user_emb, item_emb) ----------------
__global__ __launch_bounds__(256)
void k_concat(const float* __restrict__ ue, const float* __restrict__ ie,
              float* __restrict__ all, int n) {
    int i = blockIdx.x * blockDim.x + threadIdx.x;
    if (i >= n) return;
    all[i] = (i < NUM_USERS * LATENT_DIM) ? ue[i] : ie[i - NUM_USERS * LATENT_DIM];
}

// ---------------- degree = bincount(train_users) + bincount(item_nodes) ----
__global__ __launch_bounds__(256)
void k_degree(const int* __restrict__ tu, const int* __restrict__ ti,
              float* __restrict__ deg, int nE) {
    int e = blockIdx.x * blockDim.x + threadIdx.x;
    if (e >= nE) return;
    atomAddF(deg + tu[e], 1.0f);
    atomAddF(deg + NUM_USERS + ti[e], 1.0f);
}

// ---------------- COO SpMM: y[r] += v * x[c], DIM in {64,128} -------------
// G=DIM/4 consecutive threads cover one nnz -> coalesced float4 row traffic.
template <int DIM>
__global__ __launch_bounds__(256)
void k_spmm(const int* __restrict__ rows, const int* __restrict__ cols,
            const float* __restrict__ vals, const float* __restrict__ x,
            float* __restrict__ y, int nnz) {
    constexpr int G = DIM / 4;
    int idx = blockIdx.x * blockDim.x + threadIdx.x;
    if (idx >= nnz * G) return;
    int e = idx / G;
    int g = idx - e * G;
    int r = rows[e], c = cols[e];
    float v = vals[e];
    const float4 xv = *(const float4*)(x + (size_t)c * DIM + g * 4);
    float* yp = y + (size_t)r * DIM + g * 4;
    atomAddF(yp + 0, v * xv.x);
    atomAddF(yp + 1, v * xv.y);
    atomAddF(yp + 2, v * xv.z);
    atomAddF(yp + 3, v * xv.w);
}

// ---------------- feat[e] = nb[u] || nb[NU+i]  (+cat_emb on layer 0) ------
__global__ __launch_bounds__(256)
void k_feat(const float* __restrict__ nb, const int* __restrict__ tu,
            const int* __restrict__ ti, const int* __restrict__ itc,
            const float* __restrict__ cat_emb, int add_cat,
            float* __restrict__ feat, int nE) {
    int idx = blockIdx.x * blockDim.x + threadIdx.x;
    if (idx >= nE * 32) return;
    int e = idx >> 5;
    int g = idx & 31;                       // 32 float4 groups = 128 floats
    int u = tu[e], it = ti[e];
    const float* src = (g < 16)
        ? nb + (size_t)u * LATENT_DIM + g * 4
        : nb + (size_t)(NUM_USERS + it) * LATENT_DIM + (g - 16) * 4;
    float4 f = *(const float4*)src;
    if (add_cat) {
        const float4 cv = *(const float4*)(cat_emb + (size_t)itc[it] * 128 + g * 4);
        f.x += cv.x; f.y += cv.y; f.z += cv.z; f.w += cv.w;
    }
    *(float4*)(feat + (size_t)e * 128 + g * 4) = f;
}

// ---------------- scatter edge halves back to nodes -----------------------
__global__ __launch_bounds__(256)
void k_scatter(const float* __restrict__ feat2, const int* __restrict__ tu,
               const int* __restrict__ ti, float* __restrict__ upd, int nE) {
    int idx = blockIdx.x * blockDim.x + threadIdx.x;
    if (idx >= nE * 32) return;
    int e = idx >> 5;
    int g = idx & 31;
    float4 f;
    float* p;
    if (g < 16) {
        f = *(const float4*)(feat2 + (size_t)e * 128 + g * 4);
        p = upd + (size_t)tu[e] * LATENT_DIM + g * 4;
    } else {
        int gg = g - 16;
        f = *(const float4*)(feat2 + (size_t)e * 128 + 64 + gg * 4);
        p = upd + (size_t)(NUM_USERS + ti[e]) * LATENT_DIM + gg * 4;
    }
    atomAddF(p + 0, f.x);
    atomAddF(p + 1, f.y);
    atomAddF(p + 2, f.z);
    atomAddF(p + 3, f.w);
}

// ---- bufA = (bufA/(deg+eps) + bufB)*0.5 ; emb_sum += bufA ----------------
__global__ __launch_bounds__(256)
void k_combine(float* __restrict__ bufA, const float* __restrict__ bufB,
               const float* __restrict__ deg, float* __restrict__ emb_sum, int n) {
    int i = blockIdx.x * blockDim.x + threadIdx.x;
    if (i >= n) return;
    float d = deg[i >> 6];                  // LATENT_DIM == 64
    float a = (bufA[i] / (d + 1e-9f) + bufB[i]) * 0.5f;
    bufA[i] = a;
    emb_sum[i] += a;
}

// ---------------- gamma via V_WMMA_F32_16X16X4_F32 ------------------------
// One wave computes 16 dot products: C(16x16) = U(16x64) . IT(16x64)^T via
// 16 chained k=4 WMMAs; gamma = diag(C) * (1/3)^2 (layer-mean on both sides).
// A 16x4 layout: lane<16 holds M=lane K={0,1}; lane>=16 K={2,3}. B 4x16 is
// the mirror, so BOTH operands load 2 floats at offset 4*kk + 2*(lane>=16)
// from row (lane&15) of their tile.
__global__ __launch_bounds__(32)
void k_gamma(const float* __restrict__ emb_sum, const int* __restrict__ users,
             const int* __restrict__ items, float* __restrict__ out) {
    const int lane = threadIdx.x;           // 0..31, all active (EXEC = ~0)
    const int row  = lane & 15;
    const int hi   = lane >> 4;
    const int b    = blockIdx.x * 16 + row;
    const float* urow = emb_sum + (size_t)users[b] * LATENT_DIM;
    const float* irow = emb_sum + (size_t)(NUM_USERS + items[b]) * LATENT_DIM;

    v8f c = {0.f, 0.f, 0.f, 0.f, 0.f, 0.f, 0.f, 0.f};
#pragma unroll
    for (int kk = 0; kk < 16; ++kk) {
        const int kb = kk * 4 + hi * 2;
        v2f a, bb;
        a[0]  = urow[kb]; a[1]  = urow[kb + 1];
        bb[0] = irow[kb]; bb[1] = irow[kb + 1];
        c = __builtin_amdgcn_wmma_f32_16x16x4_f32(
                /*neg_a=*/false, a, /*neg_b=*/false, bb,
                /*c_mod=*/(short)0, c, /*reuse_a=*/false, /*reuse_b=*/false);
    }
    // C/D layout: VGPR v, lanes 0-15 -> M=v ; lanes 16-31 -> M=v+8 ; N=lane&15.
    // diag j<8 lives in lane j (vgpr j); diag j>=8 in lane j+16 (vgpr j-8).
    int j = -1, v = 0;
    if (lane < 8)        { j = lane;      v = lane;      }
    else if (lane >= 24) { j = lane - 16; v = lane - 24; }
    if (j >= 0) {
        float val;
        switch (v) {
            case 0: val = c[0]; break; case 1: val = c[1]; break;
            case 2: val = c[2]; break; case 3: val = c[3]; break;
            case 4: val = c[4]; break; case 5: val = c[5]; break;
            case 6: val = c[6]; break; default: val = c[7]; break;
        }
        out[blockIdx.x * 16 + j] = val * (1.0f / 9.0f);
    }
}

// ---------------------------------------------------------------------------
static inline int nblk(long long n, int b) { return (int)((n + b - 1) / b); }

extern "C" void kernel_launch(void* const* d_in, const int* in_sizes, int n_in,
                              void* d_out, int out_size, void* d_ws, size_t ws_size,
                              hipStream_t stream) {
    const float* user_emb = (const float*)d_in[0];
    const float* item_emb = (const float*)d_in[1];
    const float* cat_emb  = (const float*)d_in[2];
    const float* g_vals   = (const float*)d_in[3];
    const float* gx_vals  = (const float*)d_in[4];
    const int*   g_rows   = (const int*)d_in[5];
    const int*   g_cols   = (const int*)d_in[6];
    const int*   gx_rows  = (const int*)d_in[7];
    const int*   gx_cols  = (const int*)d_in[8];
    const int*   tr_users = (const int*)d_in[9];
    const int*   tr_items = (const int*)d_in[10];
    const int*   itc      = (const int*)d_in[11];
    const int*   users    = (const int*)d_in[12];
    const int*   items    = (const int*)d_in[13];
    float* out = (float*)d_out;

    // workspace partition (floats); total ~147.5M floats (~590 MB)
    float* ws = (float*)d_ws;
    size_t o = 0;
    float* bufA    = ws + o; o += NODE_ELEMS;   // current emb / upd target
    float* bufB    = ws + o; o += NODE_ELEMS;   // G @ bufA
    float* emb_sum = ws + o; o += NODE_ELEMS;   // sum of layer outputs
    float* degree  = ws + o; o += 131072;       // N_NODES padded
    float* feat    = ws + o; o += EDGE_FEATS;   // [E,128]
    float* feat2   = ws + o; o += EDGE_FEATS;   // GX @ feat

    const int B = 256;

    hipMemsetAsync(emb_sum, 0, (size_t)NODE_ELEMS * sizeof(float), stream);
    hipMemsetAsync(degree, 0, (size_t)N_NODES * sizeof(float), stream);
    k_concat<<<nblk(NODE_ELEMS, B), B, 0, stream>>>(user_emb, item_emb, bufA, NODE_ELEMS);
    k_degree<<<nblk(N_EDGES, B), B, 0, stream>>>(tr_users, tr_items, degree, N_EDGES);

    for (int layer = 0; layer < N_LAYERS; ++layer) {
        // bufB = G @ bufA
        hipMemsetAsync(bufB, 0, (size_t)NODE_ELEMS * sizeof(float), stream);
        k_spmm<64><<<nblk((long long)NNZ_G * 16, B), B, 0, stream>>>(
            g_rows, g_cols, g_vals, bufA, bufB, NNZ_G);
        // feat = concat(bufB[u], bufB[NU+i]) (+cat on layer 0)
        k_feat<<<nblk((long long)N_EDGES * 32, B), B, 0, stream>>>(
            bufB, tr_users, tr_items, itc, cat_emb, layer == 0 ? 1 : 0, feat, N_EDGES);
        // feat2 = GX @ feat
        hipMemsetAsync(feat2, 0, (size_t)EDGE_FEATS * sizeof(float), stream);
        k_spmm<128><<<nblk((long long)NNZ_GX * 32, B), B, 0, stream>>>(
            gx_rows, gx_cols, gx_vals, feat, feat2, NNZ_GX);
        // bufA := scatter(feat2 halves)
        hipMemsetAsync(bufA, 0, (size_t)NODE_ELEMS * sizeof(float), stream);
        k_scatter<<<nblk((long long)N_EDGES * 32, B), B, 0, stream>>>(
            feat2, tr_users, tr_items, bufA, N_EDGES);
        // bufA = (bufA/deg + bufB)*0.5 ; emb_sum += bufA
        k_combine<<<nblk(NODE_ELEMS, B), B, 0, stream>>>(
            bufA, bufB, degree, emb_sum, NODE_ELEMS);
    }

    // gamma = diag(U . IT^T) / 9 via f32 WMMA; one wave per 16 batch rows
    k_gamma<<<BATCH_SZ / 16, 32, 0, stream>>>(emb_sum, users, items, out);
}